// HyperbolicAttentionLayer_10728828305786
// MI455X (gfx1250) — compile-verified
//
#include <hip/hip_runtime.h>
#include <math.h>

#define N_NODES 50000
#define N_EDGES 800000
#define DIM     128
#define NHEADS  4
#define EDIM    32
#define EPS_    1e-8f

#define KPAD    132           // 129 rounded up to multiple of 4 (+ zero pad)
#define TILES   25            // 16-row tiles per GEMM block (125*25 = 3125)

typedef __attribute__((ext_vector_type(2))) float v2f;
typedef __attribute__((ext_vector_type(8))) float v8f;

__device__ __forceinline__ float waveReduceSum(float v) {
#pragma unroll
  for (int off = 16; off > 0; off >>= 1) v += __shfl_xor(v, off, 32);
  return v;
}

// float atomic max via ordered-int trick (init must be -inf)
__device__ __forceinline__ void atomicMaxF(float* addr, float val) {
  if (val >= 0.0f) atomicMax((int*)addr, __float_as_int(val));
  else             atomicMin((unsigned int*)addr, __float_as_uint(val));
}

__global__ void fill_kernel(float* __restrict__ p, float v, size_t n) {
  size_t i = (size_t)blockIdx.x * blockDim.x + threadIdx.x;
  size_t stride = (size_t)gridDim.x * blockDim.x;
  for (; i < n; i += stride) p[i] = v;
}

// ---------------------------------------------------------------------------
// Phase 1: space = x @ W^T (+bias), fp32 WMMA 16x16x4.
// Block = 256 threads = 8 waves. The 128x129 weight slice for this
// (head, q/k/v) is staged once into LDS (66KB, padded to KPAD stride so the
// zero pad makes the K tail guard-free), then the block sweeps TILES=25
// 16-row tiles. Inner loop: ds_load(B) + global_load(A) + v_wmma, no
// branches, EXEC all-1s throughout.
// Output layout: out[node][head][128] contiguous rows.
// ---------------------------------------------------------------------------
__global__ __launch_bounds__(256) void hyp_gemm_wmma(
    const float* __restrict__ x,                                  // (N,129)
    const float* __restrict__ Wq, const float* __restrict__ Wk,
    const float* __restrict__ Wv,                                 // (H,128,129)
    const float* __restrict__ bq, const float* __restrict__ bk,
    const float* __restrict__ bv,                                 // (H,128)
    float* __restrict__ qs, float* __restrict__ ks,
    float* __restrict__ vs)                                       // (N,H,128)
{
  __shared__ float Blds[DIM * KPAD];                              // 67.6 KB

  const int h     = blockIdx.y;
  const int which = blockIdx.z;
  const float* W    = (which == 0) ? Wq : (which == 1) ? Wk : Wv;
  const float* bias = (which == 0) ? bq : (which == 1) ? bk : bv;
  float* out        = (which == 0) ? qs : (which == 1) ? ks : vs;

  // --- stage B = W[h] (transposed access pattern) into LDS, zero the pad ---
  const float* Wh = W + (size_t)h * DIM * 129;
  for (int idx = threadIdx.x; idx < DIM * 129; idx += 256) {
    const int c  = idx / 129;
    const int kk = idx - c * 129;
    Blds[c * KPAD + kk] = Wh[idx];
  }
  for (int idx = threadIdx.x; idx < DIM * (KPAD - 129); idx += 256) {
    const int c = idx / (KPAD - 129);
    Blds[c * KPAD + 129 + (idx - c * (KPAD - 129))] = 0.0f;
  }
  __syncthreads();

  const int wave = threadIdx.x >> 5;
  const int lane = threadIdx.x & 31;
  const int mrow = lane & 15;
  const int hi   = lane >> 4;            // 0: K pair {0,1}, 1: K pair {2,3}
  const int col  = wave * 16 + mrow;     // output column 0..127

  const float* __restrict__ Bcol = Blds + col * KPAD;
  const float  bb = bias[h * DIM + col];

  for (int t = 0; t < TILES; ++t) {
    const int n0 = (blockIdx.x * TILES + t) * 16;
    const float* __restrict__ xrow = x + (size_t)(n0 + mrow) * 129;

    v8f acc = {0.f, 0.f, 0.f, 0.f, 0.f, 0.f, 0.f, 0.f};
#pragma unroll 4
    for (int k = 0; k < 128; k += 4) {   // kk+1 <= 127 < 129: no guards
      const int kk = k + 2 * hi;
      v2f a, b;
      a.x = xrow[kk];  a.y = xrow[kk + 1];
      b.x = Bcol[kk];  b.y = Bcol[kk + 1];
      acc = __builtin_amdgcn_wmma_f32_16x16x4_f32(
          false, a, false, b, (short)0, acc, false, false);
    }
    {                                    // K tail: element 128 only (hi==0 slot)
      const float xt = xrow[128];        // in-bounds for every lane; select below
      v2f a, b;
      a.x = hi ? 0.0f : xt;              // v_cndmask, no divergence
      a.y = 0.0f;
      b.x = Bcol[128 + 2 * hi];          // hi=1 reads zero pad
      b.y = Bcol[129 + 2 * hi];          // zero pad
      acc = __builtin_amdgcn_wmma_f32_16x16x4_f32(
          false, a, false, b, (short)0, acc, false, false);
    }

#pragma unroll
    for (int r = 0; r < 8; ++r) {
      const int row = n0 + 8 * hi + r;   // C/D layout: lanes16-31 hold M=8..15
      out[((size_t)row * NHEADS + h) * DIM + col] = acc[r] + bb;
    }
  }
}

// ---------------------------------------------------------------------------
// Phase 1b: per (node,head) epilogue. z=0/1: time = sqrt(|s|^2+K) for q/k.
// z=2: rewrite v-space in place as tan = logmap0_space(v).
// One wave per (node,head) unit; 4 elems per lane.
// ---------------------------------------------------------------------------
__global__ __launch_bounds__(256) void node_epilogue(
    float* __restrict__ qs, float* __restrict__ ks, float* __restrict__ vs,
    float* __restrict__ qt, float* __restrict__ kt)
{
  const int unit = blockIdx.x * 8 + (threadIdx.x >> 5);
  if (unit >= N_NODES * NHEADS) return;
  const int lane  = threadIdx.x & 31;
  const int which = blockIdx.z;
  float* row = ((which == 0) ? qs : (which == 1) ? ks : vs) + (size_t)unit * DIM;

  float v0 = row[lane], v1 = row[lane + 32], v2 = row[lane + 64], v3 = row[lane + 96];
  float ss = waveReduceSum(v0 * v0 + v1 * v1 + v2 * v2 + v3 * v3);
  if (which < 2) {
    if (lane == 0) ((which == 0) ? qt : kt)[unit] = sqrtf(ss + 1.0f);
  } else {
    float ns   = sqrtf(ss);
    float time = sqrtf(ss + 1.0f);
    float dist = acoshf(fmaxf(time, 1.0f));
    float sc   = dist / fmaxf(ns, EPS_);
    row[lane] = v0 * sc; row[lane + 32] = v1 * sc;
    row[lane + 64] = v2 * sc; row[lane + 96] = v3 * sc;
  }
}

// ---------------------------------------------------------------------------
// Phase 2a: per-edge logits + segment max. One wave per edge.
// ---------------------------------------------------------------------------
__global__ __launch_bounds__(256) void edge_logits(
    const int* __restrict__ src, const int* __restrict__ dst,
    const float* __restrict__ ef, const float* __restrict__ We,
    const float* __restrict__ qs, const float* __restrict__ qt,
    const float* __restrict__ ks, const float* __restrict__ kt,
    float* __restrict__ logits, float* __restrict__ mbuf)
{
  const int e = blockIdx.x * 8 + (threadIdx.x >> 5);
  if (e >= N_EDGES) return;
  const int lane = threadIdx.x & 31;
  const int s = src[e], d = dst[e];
  const float efv = ef[(size_t)e * EDIM + lane];      // EDIM == 32 == wave
  const float inv_scale = 0.08838834764831845f;       // 1/sqrt(128)
#pragma unroll
  for (int h = 0; h < NHEADS; ++h) {
    const float* qrow = qs + ((size_t)d * NHEADS + h) * DIM;
    const float* krow = ks + ((size_t)s * NHEADS + h) * DIM;
    float dot = qrow[lane] * krow[lane] + qrow[lane + 32] * krow[lane + 32]
              + qrow[lane + 64] * krow[lane + 64] + qrow[lane + 96] * krow[lane + 96];
    dot = waveReduceSum(dot);
    float ew = waveReduceSum(efv * We[h * EDIM + lane]);
    if (lane == 0) {
      float li = -qt[d * NHEADS + h] * kt[s * NHEADS + h] + dot;
      float lg = (2.0f + 2.0f * li) * inv_scale + ew;
      logits[(size_t)e * NHEADS + h] = lg;
      atomicMaxF(&mbuf[d * NHEADS + h], lg);
    }
  }
}

// ---------------------------------------------------------------------------
// Phase 2b: ex = exp(logit - m[dst]); den += ex. One thread per (edge,head).
// ---------------------------------------------------------------------------
__global__ void edge_softmax_num(
    const int* __restrict__ dst, const float* __restrict__ mbuf,
    float* __restrict__ logits, float* __restrict__ den)
{
  const size_t idx = (size_t)blockIdx.x * blockDim.x + threadIdx.x;
  if (idx >= (size_t)N_EDGES * NHEADS) return;
  const int e = (int)(idx >> 2);
  const int h = (int)(idx & 3);
  const int d = dst[e];
  const float ex = __expf(logits[idx] - mbuf[d * NHEADS + h]);
  logits[idx] = ex;
  atomicAdd(&den[d * NHEADS + h], ex);
}

// ---------------------------------------------------------------------------
// Phase 2c: agg[dst] += alpha * tan[src]. One wave per edge, coalesced
// 512B row gathers, fp32 scatter atomics.
// ---------------------------------------------------------------------------
__global__ __launch_bounds__(256) void edge_aggregate(
    const int* __restrict__ src, const int* __restrict__ dst,
    const float* __restrict__ ex, const float* __restrict__ den,
    const float* __restrict__ tanv, float* __restrict__ agg)
{
  const int e = blockIdx.x * 8 + (threadIdx.x >> 5);
  if (e >= N_EDGES) return;
  const int lane = threadIdx.x & 31;
  const int s = src[e], d = dst[e];
#pragma unroll
  for (int h = 0; h < NHEADS; ++h) {
    const float alpha = ex[(size_t)e * NHEADS + h] / den[d * NHEADS + h];
    const float* trow = tanv + ((size_t)s * NHEADS + h) * DIM;
    float* arow = agg + ((size_t)d * NHEADS + h) * DIM;
    atomicAdd(&arow[lane],      alpha * trow[lane]);
    atomicAdd(&arow[lane + 32], alpha * trow[lane + 32]);
    atomicAdd(&arow[lane + 64], alpha * trow[lane + 64]);
    atomicAdd(&arow[lane + 96], alpha * trow[lane + 96]);
  }
}

// ---------------------------------------------------------------------------
// Phase 3: head_pts = expmap0(agg); mean of logmap0_space over heads;
// out = expmap0(mean). One wave per node; writes all 129 outputs.
// ---------------------------------------------------------------------------
__global__ __launch_bounds__(256) void node_output(
    const float* __restrict__ agg, float* __restrict__ out)
{
  const int n = blockIdx.x * 8 + (threadIdx.x >> 5);
  if (n >= N_NODES) return;
  const int lane = threadIdx.x & 31;
  float m0 = 0.f, m1 = 0.f, m2 = 0.f, m3 = 0.f;
#pragma unroll
  for (int h = 0; h < NHEADS; ++h) {
    const float* u = agg + ((size_t)n * NHEADS + h) * DIM;
    float u0 = u[lane], u1 = u[lane + 32], u2 = u[lane + 64], u3 = u[lane + 96];
    float ss  = waveReduceSum(u0 * u0 + u1 * u1 + u2 * u2 + u3 * u3);
    float nrm = sqrtf(ss);
    // expmap0: time=cosh(nrm), space = sinh(nrm)/max(nrm,eps) * u
    float time = coshf(nrm);
    float sc   = sinhf(nrm) / fmaxf(nrm, EPS_);
    // logmap0_space: ns = |sc|*nrm (exact), dist = acosh(max(time,1))
    float ns   = fabsf(sc) * nrm;
    float dist = acoshf(fmaxf(time, 1.0f));
    float tsc  = (dist / fmaxf(ns, EPS_)) * sc * 0.25f;   // 0.25 = mean over 4 heads
    m0 += tsc * u0; m1 += tsc * u1; m2 += tsc * u2; m3 += tsc * u3;
  }
  float ssm = waveReduceSum(m0 * m0 + m1 * m1 + m2 * m2 + m3 * m3);
  float nm  = sqrtf(ssm);
  float otime = coshf(nm);
  float osc   = sinhf(nm) / fmaxf(nm, EPS_);
  float* orow = out + (size_t)n * 129;
  if (lane == 0) orow[0] = otime;
  orow[1 + lane]      = osc * m0;
  orow[1 + lane + 32] = osc * m1;
  orow[1 + lane + 64] = osc * m2;
  orow[1 + lane + 96] = osc * m3;
}

extern "C" void kernel_launch(void* const* d_in, const int* in_sizes, int n_in,
                              void* d_out, int out_size, void* d_ws, size_t ws_size,
                              hipStream_t stream) {
  const float* x  = (const float*)d_in[0];
  const int*   ei = (const int*)d_in[1];        // (2,E)
  const float* ef = (const float*)d_in[2];
  const float* Wq = (const float*)d_in[3];
  const float* bq = (const float*)d_in[4];
  const float* Wk = (const float*)d_in[5];
  const float* bk = (const float*)d_in[6];
  const float* Wv = (const float*)d_in[7];
  const float* bv = (const float*)d_in[8];
  const float* We = (const float*)d_in[9];
  float* out = (float*)d_out;
  const int* src = ei;
  const int* dst = ei + N_EDGES;

  // Workspace carve-up (floats): ~425 MB total
  float* w = (float*)d_ws;
  const size_t NH  = (size_t)N_NODES * NHEADS;   // 200k
  const size_t NHD = NH * DIM;                   // 25.6M
  float* qs     = w; w += NHD;
  float* ks     = w; w += NHD;
  float* tanv   = w; w += NHD;
  float* aggb   = w; w += NHD;
  float* qt     = w; w += NH;
  float* kt     = w; w += NH;
  float* mbuf   = w; w += NH;
  float* den    = w; w += NH;
  float* logits = w; w += (size_t)N_EDGES * NHEADS;

  fill_kernel<<<4096, 256, 0, stream>>>(aggb, 0.0f, NHD);
  fill_kernel<<<512, 256, 0, stream>>>(den, 0.0f, NH);
  fill_kernel<<<512, 256, 0, stream>>>(mbuf, -INFINITY, NH);

  dim3 g1(N_NODES / 16 / TILES, NHEADS, 3);      // 125 x 4 x 3
  hyp_gemm_wmma<<<g1, 256, 0, stream>>>(x, Wq, Wk, Wv, bq, bk, bv, qs, ks, tanv);

  dim3 g2((unsigned)((NH + 7) / 8), 1, 3);
  node_epilogue<<<g2, 256, 0, stream>>>(qs, ks, tanv, qt, kt);

  edge_logits<<<(N_EDGES + 7) / 8, 256, 0, stream>>>(src, dst, ef, We, qs, qt,
                                                     ks, kt, logits, mbuf);

  edge_softmax_num<<<(unsigned)(((size_t)N_EDGES * NHEADS + 255) / 256), 256, 0, stream>>>(
      dst, mbuf, logits, den);

  edge_aggregate<<<(N_EDGES + 7) / 8, 256, 0, stream>>>(src, dst, logits, den,
                                                        tanv, aggb);

  node_output<<<(N_NODES + 7) / 8, 256, 0, stream>>>(aggb, out);
}